// metapath_mtne_29137058136494
// MI455X (gfx1250) — compile-verified
//
#include <hip/hip_runtime.h>
#include <math.h>

#define DD  128   // embedding dim
#define MM  4     // metapaths
#define EE  10    // edges per metapath
#define QQ  5     // negative samples
#define MPTT 8    // metapath types

typedef float v2f __attribute__((ext_vector_type(2)));
typedef float v8f __attribute__((ext_vector_type(8)));

// ---------------------------------------------------------------------------
// Kernel 1: embW = emb @ W   ([N,128] x [128,128] fp32 GEMM via WMMA f32 K=4)
// Block = 256 threads = 8 waves; each wave computes a 16-row strip.
// W (64KB) staged in LDS once per block.
// ---------------------------------------------------------------------------
__global__ __launch_bounds__(256) void embw_gemm(const float* __restrict__ emb,
                                                 const float* __restrict__ W,
                                                 float* __restrict__ embW,
                                                 int nrows) {
  __shared__ float Wl[DD * DD];  // 64 KB
  for (int i = threadIdx.x; i < DD * DD / 4; i += 256)
    ((float4*)Wl)[i] = ((const float4*)W)[i];
  __syncthreads();

  const int lane = threadIdx.x & 31;
  const int wave = threadIdx.x >> 5;
  const int r0 = (blockIdx.x * 8 + wave) * 16;
  if (r0 >= nrows) return;  // wave-uniform exit (EXEC stays all-1 for WMMA)

  const int half = lane >> 4;   // 0: K={0,1}, 1: K={2,3} within each K-step
  const int l15  = lane & 15;
  const int arow = min(r0 + l15, nrows - 1);  // clamp tail (stores masked below)

  // A fragments: 16x4 f32 tile per K-step. Lane l<16 -> row l, K=k0+{0,1};
  // lane l>=16 -> row l-16, K=k0+{2,3}.  32 K-steps cover K=128.
  v2f afrag[32];
#pragma unroll
  for (int ks = 0; ks < 32; ++ks) {
    const int k = 4 * ks + 2 * half;
    afrag[ks] = *(const v2f*)(emb + (long)arow * DD + k);
  }

  for (int nt = 0; nt < 8; ++nt) {  // 8 column tiles of 16
    const int n0 = nt * 16;
    v8f c = {0.f, 0.f, 0.f, 0.f, 0.f, 0.f, 0.f, 0.f};
#pragma unroll
    for (int ks = 0; ks < 32; ++ks) {
      const int k = 4 * ks + 2 * half;
      // B fragment: 4x16 tile, lane = column, lane-half selects K-pair.
      v2f bfrag;
      bfrag.x = Wl[k * DD + n0 + l15];
      bfrag.y = Wl[(k + 1) * DD + n0 + l15];
      c = __builtin_amdgcn_wmma_f32_16x16x4_f32(false, afrag[ks], false, bfrag,
                                                (short)0, c, false, false);
    }
    // C/D layout: VGPR v, lanes 0-15 -> M=v, lanes 16-31 -> M=v+8.
#pragma unroll
    for (int v = 0; v < 8; ++v) {
      const int row = r0 + v + 8 * half;
      if (row < nrows) embW[(long)row * DD + n0 + l15] = c[v];
    }
  }
}

// ---------------------------------------------------------------------------
// Kernel 2: per-batch pair dots + decay/attention weighting + loss.
// One block per batch element b; 246 dot-tasks spread over 8 waves.
// ---------------------------------------------------------------------------
__global__ __launch_bounds__(256) void pair_loss(
    const float* __restrict__ emb, const float* __restrict__ delta,
    const float* __restrict__ distance_att, const float* __restrict__ metapath_att,
    const int* __restrict__ s_node, const int* __restrict__ t_node,
    const float* __restrict__ s_t_time,
    const int* __restrict__ mp_src, const int* __restrict__ mp_dst,
    const float* __restrict__ mp_time, const int* __restrict__ mp_dist,
    const int* __restrict__ mp_type,
    const int* __restrict__ neg_node,
    const int* __restrict__ neg_mp_src, const int* __restrict__ neg_mp_dst,
    const float* __restrict__ neg_mp_time, const int* __restrict__ neg_mp_dist,
    const int* __restrict__ neg_mp_type,
    const float* __restrict__ embW, float* __restrict__ loss) {
  __shared__ float s_dis[EE];
  __shared__ float s_mp[MPTT];
  __shared__ float s_acc[1 + QQ];  // [0]=p_lambda acc, [1+q]=neg acc

  const int b = blockIdx.x;

  if (threadIdx.x == 0) {
    // softmax(distance_att), softmax(metapath_att) -- tiny, done serially
    float mx = -INFINITY;
    for (int i = 0; i < EE; ++i) mx = fmaxf(mx, distance_att[i]);
    float s = 0.f;
    for (int i = 0; i < EE; ++i) { float e = expf(distance_att[i] - mx); s_dis[i] = e; s += e; }
    for (int i = 0; i < EE; ++i) s_dis[i] /= s;
    mx = -INFINITY;
    for (int i = 0; i < MPTT; ++i) mx = fmaxf(mx, metapath_att[i]);
    s = 0.f;
    for (int i = 0; i < MPTT; ++i) { float e = expf(metapath_att[i] - mx); s_mp[i] = e; s += e; }
    for (int i = 0; i < MPTT; ++i) s_mp[i] /= s;
    for (int i = 0; i <= QQ; ++i) s_acc[i] = 0.f;
  }
  __syncthreads();

  const int lane = threadIdx.x & 31;
  const int wave = threadIdx.x >> 5;

  const int   sn = s_node[b];
  const float t0 = s_t_time[b];
  const float ds = delta[sn];

  const int NTASK = QQ * MM * EE + MM * EE + QQ + 1;  // 200 + 40 + 5 + 1 = 246
  for (int t = wave; t < NTASK; t += 8) {
    int src, dst, slot;
    float w;
    if (t < QQ * MM * EE) {                       // negative metapath edge
      const int q = t / (MM * EE);
      const int r = t - q * (MM * EE);
      const int m = r / EE;
      const int e = r - m * EE;
      const int base = ((b * QQ + q) * MM + m) * EE + e;
      src = neg_mp_src[base];
      dst = neg_mp_dst[base];
      const float tm = neg_mp_time[base];
      const int   dd = neg_mp_dist[base];
      const int   ty = neg_mp_type[(b * QQ + q) * MM + m];
      const float dq = delta[neg_node[b * QQ + q]];
      w = s_dis[dd] * expf(-ds * dq * fabsf(t0 - tm)) * s_mp[ty];
      slot = 1 + q;
    } else if (t < QQ * MM * EE + MM * EE) {      // positive metapath edge
      const int r = t - QQ * MM * EE;
      const int m = r / EE;
      const int e = r - m * EE;
      const int base = (b * MM + m) * EE + e;
      src = mp_src[base];
      dst = mp_dst[base];
      const float tm = mp_time[base];
      const int   dd = mp_dist[base];
      const int   ty = mp_type[b * MM + m];
      const float dt = delta[t_node[b]];
      w = s_dis[dd] * expf(-ds * dt * fabsf(t0 - tm)) * s_mp[ty];
      slot = 0;
    } else if (t < QQ * MM * EE + MM * EE + QQ) { // s vs neg-node direct sim
      const int q = t - (QQ * MM * EE + MM * EE);
      src = sn;
      dst = neg_node[b * QQ + q];
      w = 1.f;
      slot = 1 + q;
    } else {                                      // s vs t direct sim
      src = sn;
      dst = t_node[b];
      w = 1.f;
      slot = 0;
    }

    // 128-dot: float4 per lane, then 5-step wave32 xor reduction
    const float4 av = *(const float4*)(embW + (long)src * DD + lane * 4);
    const float4 bv = *(const float4*)(emb + (long)dst * DD + lane * 4);
    float p = av.x * bv.x + av.y * bv.y + av.z * bv.z + av.w * bv.w;
#pragma unroll
    for (int off = 16; off > 0; off >>= 1) p += __shfl_xor(p, off, 32);
    if (lane == 0) atomicAdd(&s_acc[slot], p * w);  // ds_add_f32
  }
  __syncthreads();

  if (threadIdx.x == 0) {
    const float eps = 1e-6f;
    const float pl = s_acc[0];
    float nl = 0.f;
    for (int q = 0; q < QQ; ++q) {
      const float x  = -s_acc[1 + q];
      const float sg = 1.f / (1.f + expf(-x));
      nl += logf(sg + eps);
    }
    const float sp = 1.f / (1.f + expf(-pl));
    loss[b] = -logf(sp + eps) - nl;
  }
}

// ---------------------------------------------------------------------------
extern "C" void kernel_launch(void* const* d_in, const int* in_sizes, int n_in,
                              void* d_out, int out_size, void* d_ws, size_t ws_size,
                              hipStream_t stream) {
  const float* emb          = (const float*)d_in[0];
  const float* delta        = (const float*)d_in[1];
  const float* W            = (const float*)d_in[2];
  const float* distance_att = (const float*)d_in[3];
  const float* metapath_att = (const float*)d_in[4];
  const int*   s_node       = (const int*)d_in[5];
  const int*   t_node       = (const int*)d_in[6];
  const float* s_t_time     = (const float*)d_in[7];
  const int*   mp_src       = (const int*)d_in[8];
  const int*   mp_dst       = (const int*)d_in[9];
  const float* mp_time      = (const float*)d_in[10];
  const int*   mp_dist      = (const int*)d_in[11];
  const int*   mp_type      = (const int*)d_in[12];
  const int*   neg_node     = (const int*)d_in[13];
  const int*   neg_mp_src   = (const int*)d_in[14];
  const int*   neg_mp_dst   = (const int*)d_in[15];
  const float* neg_mp_time  = (const float*)d_in[16];
  const int*   neg_mp_dist  = (const int*)d_in[17];
  const int*   neg_mp_type  = (const int*)d_in[18];

  const int N = in_sizes[0] / DD;   // 100000
  const int B = in_sizes[5];        // 512

  float* embW = (float*)d_ws;       // needs N*DD*4 = 51.2 MB of scratch

  embw_gemm<<<dim3((N + 127) / 128), 256, 0, stream>>>(emb, W, embW, N);
  pair_loss<<<dim3(B), 256, 0, stream>>>(emb, delta, distance_att, metapath_att,
                                         s_node, t_node, s_t_time,
                                         mp_src, mp_dst, mp_time, mp_dist, mp_type,
                                         neg_node, neg_mp_src, neg_mp_dst,
                                         neg_mp_time, neg_mp_dist, neg_mp_type,
                                         embW, (float*)d_out);
}